// Attentioncell_56684978372953
// MI455X (gfx1250) — compile-verified
//
#include <hip/hip_runtime.h>
#include <hip/hip_bf16.h>

// Shapes from the reference
#define Bn 32
#define Tn 32
#define Ln 196
#define Sn 512
#define Dn 512

typedef __attribute__((ext_vector_type(2))) float v2f;
typedef __attribute__((ext_vector_type(8))) float v8f;

__device__ __forceinline__ float wave_reduce_sum(float v) {
  #pragma unroll
  for (int off = 16; off > 0; off >>= 1) v += __shfl_xor(v, off, 32);
  return v;
}
__device__ __forceinline__ float wave_reduce_max(float v) {
  #pragma unroll
  for (int off = 16; off > 0; off >>= 1) v = fmaxf(v, __shfl_xor(v, off, 32));
  return v;
}

// ---------------------------------------------------------------------------
// Kernel A: w1v[s] = dot(W1[s,:], V)   (512 x 512 GEMV)
// grid 16 x 256 threads -> 128 waves, 4 rows per wave
// ---------------------------------------------------------------------------
__global__ void k_w1v(const float* __restrict__ W1, const float* __restrict__ V,
                      float* __restrict__ w1v) {
  const int gwave = (blockIdx.x * blockDim.x + threadIdx.x) >> 5;  // 0..127
  const int lane  = threadIdx.x & 31;
  #pragma unroll
  for (int i = 0; i < 4; ++i) {
    const int s = gwave * 4 + i;  // 0..511
    const float* row = W1 + (size_t)s * Dn;
    float acc = 0.f;
    #pragma unroll
    for (int it = 0; it < 4; ++it) {
      const int d = it * 128 + lane * 4;
      const float4 w  = *(const float4*)(row + d);
      const float4 vv = *(const float4*)(V + d);
      acc += w.x * vv.x + w.y * vv.y + w.z * vv.z + w.w * vv.w;
    }
    acc = wave_reduce_sum(acc);
    if (lane == 0) w1v[s] = acc;
  }
}

// ---------------------------------------------------------------------------
// Kernel B: per batch b (softmax shift-invariance kills h@W2@V and b2.V):
//   scores[l] = x_static[b,l,:] . w1v
//   attn      = softmax(scores)
//   ctx[b,s]  = sum_l attn[l] * x_static[b,l,s]
// grid 32 blocks (one per b) x 256 threads (8 waves)
// ---------------------------------------------------------------------------
__global__ void k_attn_ctx(const float* __restrict__ xs, const float* __restrict__ w1v,
                           float* __restrict__ ctx) {
  __shared__ float s_w1v[Dn];
  __shared__ float s_sc[Ln + 4];
  __shared__ float s_red[2];
  const int b    = blockIdx.x;
  const int t    = threadIdx.x;
  const int lane = t & 31;
  const int wave = t >> 5;

  s_w1v[t]       = w1v[t];
  s_w1v[t + 256] = w1v[t + 256];
  __syncthreads();

  const float* xb = xs + (size_t)b * Ln * Sn;

  for (int l = wave; l < Ln; l += 8) {
    const float* row = xb + (size_t)l * Sn;
    float acc = 0.f;
    #pragma unroll
    for (int it = 0; it < 4; ++it) {
      const int d = it * 128 + lane * 4;
      const float4 xv = *(const float4*)(row + d);
      acc += xv.x * s_w1v[d] + xv.y * s_w1v[d + 1] + xv.z * s_w1v[d + 2] + xv.w * s_w1v[d + 3];
    }
    acc = wave_reduce_sum(acc);
    if (lane == 0) s_sc[l] = acc;
  }
  __syncthreads();

  if (wave == 0) {
    float m = -3.4e38f;
    for (int l = lane; l < Ln; l += 32) m = fmaxf(m, s_sc[l]);
    m = wave_reduce_max(m);
    float ssum = 0.f;
    for (int l = lane; l < Ln; l += 32) ssum += __expf(s_sc[l] - m);
    ssum = wave_reduce_sum(ssum);
    if (lane == 0) { s_red[0] = m; s_red[1] = ssum; }
  }
  __syncthreads();
  {
    const float m = s_red[0], inv = 1.f / s_red[1];
    if (t < Ln) s_sc[t] = __expf(s_sc[t] - m) * inv;
  }
  __syncthreads();

  for (int s = t; s < Sn; s += 256) {
    float acc = 0.f;
    for (int l = 0; l < Ln; ++l) acc += s_sc[l] * xb[(size_t)l * Sn + s];
    ctx[(size_t)b * Sn + s] = acc;
  }
}

// ---------------------------------------------------------------------------
// Kernel B2: cvec[b, n] = ctx[b,:] @ W3[512:,:] + b3[n]
// 32x512x512 f32 WMMA GEMM. Wave = 16x64 strip. grid (2, 2) x 128 threads.
// ---------------------------------------------------------------------------
__global__ void k_cvec(const float* __restrict__ ctx, const float* __restrict__ W3,
                       const float* __restrict__ b3, float* __restrict__ cvec) {
  const int lane = threadIdx.x & 31;
  const int wave = threadIdx.x >> 5;
  const int m0   = blockIdx.x * 16;               // 0 or 16
  const int n0   = blockIdx.y * 256 + wave * 64;
  const int l15  = lane & 15;
  const int hi   = lane >> 4;

  const float* crow = ctx + (size_t)(m0 + l15) * Sn;
  const float* Wb   = W3 + (size_t)Dn * Dn;       // bottom half of W3

  v8f acc[4];
  #pragma unroll
  for (int j = 0; j < 4; ++j) {
    const float bias = b3[n0 + j * 16 + l15];
    #pragma unroll
    for (int r = 0; r < 8; ++r) acc[j][r] = bias;
  }

  for (int k = 0; k < Sn; k += 4) {
    const int kk = k + hi * 2;
    const float2 av = *(const float2*)(crow + kk);
    v2f a; a[0] = av.x; a[1] = av.y;
    #pragma unroll
    for (int j = 0; j < 4; ++j) {
      const float* bp = Wb + (size_t)kk * Dn + n0 + j * 16 + l15;
      v2f bfrag; bfrag[0] = bp[0]; bfrag[1] = bp[Dn];
      acc[j] = __builtin_amdgcn_wmma_f32_16x16x4_f32(
          false, a, false, bfrag, (short)0, acc[j], false, false);
    }
  }

  #pragma unroll
  for (int j = 0; j < 4; ++j) {
    const int n = n0 + j * 16 + l15;
    #pragma unroll
    for (int r = 0; r < 8; ++r)
      cvec[(size_t)(m0 + r + hi * 8) * Dn + n] = acc[j][r];
  }
}

// ---------------------------------------------------------------------------
// Kernel C (split): out[m, n] = x[m,:] @ W3[:512,:] + cvec[m>>5, n]
// K = 512 only; all 16 rows of a strip share one batch b = m0>>5.
// grid (64, 2) x 128 threads (4 waves, each a 16x64 strip).
// ---------------------------------------------------------------------------
__global__ void k_out_gemm_x(const float* __restrict__ x, const float* __restrict__ W3,
                             const float* __restrict__ cvec, float* __restrict__ out) {
  const int lane = threadIdx.x & 31;
  const int wave = threadIdx.x >> 5;
  const int m0   = blockIdx.x * 16;
  const int n0   = blockIdx.y * 256 + wave * 64;
  const int l15  = lane & 15;
  const int hi   = lane >> 4;

  const float* xrow = x + (size_t)(m0 + l15) * Dn;
  const float* cv   = cvec + (size_t)(m0 >> 5) * Dn;  // same b for whole strip

  v8f acc[4];
  #pragma unroll
  for (int j = 0; j < 4; ++j) {
    const float bias = cv[n0 + j * 16 + l15];
    #pragma unroll
    for (int r = 0; r < 8; ++r) acc[j][r] = bias;
  }

  for (int k = 0; k < Dn; k += 4) {
    const int kk = k + hi * 2;
    const float2 av = *(const float2*)(xrow + kk);
    v2f a; a[0] = av.x; a[1] = av.y;
    #pragma unroll
    for (int j = 0; j < 4; ++j) {
      const float* bp = W3 + (size_t)kk * Dn + n0 + j * 16 + l15;
      v2f bfrag; bfrag[0] = bp[0]; bfrag[1] = bp[Dn];
      acc[j] = __builtin_amdgcn_wmma_f32_16x16x4_f32(
          false, a, false, bfrag, (short)0, acc[j], false, false);
    }
  }

  #pragma unroll
  for (int j = 0; j < 4; ++j) {
    const int n = n0 + j * 16 + l15;
    #pragma unroll
    for (int r = 0; r < 8; ++r)
      out[(size_t)(m0 + r + hi * 8) * Dn + n] = acc[j][r];
  }
}

// ---------------------------------------------------------------------------
// Fallback fused kernel (K = 1024, ctx appended per-row) — used only if the
// workspace is too small for the split path.
// ---------------------------------------------------------------------------
__global__ void k_out_gemm_fused(const float* __restrict__ x, const float* __restrict__ ctx,
                                 const float* __restrict__ W3, const float* __restrict__ b3,
                                 float* __restrict__ out) {
  const int lane = threadIdx.x & 31;
  const int wave = threadIdx.x >> 5;
  const int m0   = blockIdx.x * 16;
  const int n0   = blockIdx.y * 256 + wave * 64;
  const int l15  = lane & 15;
  const int hi   = lane >> 4;

  const int rowA = m0 + l15;
  const float* xrow = x   + (size_t)rowA * Dn;
  const float* crow = ctx + (size_t)(rowA >> 5) * Sn;

  v8f acc[4];
  #pragma unroll
  for (int j = 0; j < 4; ++j) {
    const float bias = b3[n0 + j * 16 + l15];
    #pragma unroll
    for (int r = 0; r < 8; ++r) acc[j][r] = bias;
  }

  for (int k = 0; k < Dn + Sn; k += 4) {
    const int kk = k + hi * 2;
    v2f a;
    if (k < Dn) {
      const float2 av = *(const float2*)(xrow + kk);
      a[0] = av.x; a[1] = av.y;
    } else {
      const float2 av = *(const float2*)(crow + (kk - Dn));
      a[0] = av.x; a[1] = av.y;
    }
    #pragma unroll
    for (int j = 0; j < 4; ++j) {
      const float* bp = W3 + (size_t)kk * Dn + n0 + j * 16 + l15;
      v2f bfrag; bfrag[0] = bp[0]; bfrag[1] = bp[Dn];
      acc[j] = __builtin_amdgcn_wmma_f32_16x16x4_f32(
          false, a, false, bfrag, (short)0, acc[j], false, false);
    }
  }

  #pragma unroll
  for (int j = 0; j < 4; ++j) {
    const int n = n0 + j * 16 + l15;
    #pragma unroll
    for (int r = 0; r < 8; ++r)
      out[(size_t)(m0 + r + hi * 8) * Dn + n] = acc[j][r];
  }
}

// ---------------------------------------------------------------------------
// Inputs (setup_inputs order):
// 0:x [B,T,D]  1:x_static [B,L,S]  2:h0 (dead)  3:W1 [S,D]  4:W2 (dead)
// 5:W3 [D+S,D] 6:b2 (dead)         7:b3 [D]     8:V [D,1]
// ---------------------------------------------------------------------------
extern "C" void kernel_launch(void* const* d_in, const int* in_sizes, int n_in,
                              void* d_out, int out_size, void* d_ws, size_t ws_size,
                              hipStream_t stream) {
  const float* x        = (const float*)d_in[0];
  const float* x_static = (const float*)d_in[1];
  const float* W1       = (const float*)d_in[3];
  const float* W3       = (const float*)d_in[5];
  const float* b3       = (const float*)d_in[7];
  const float* V        = (const float*)d_in[8];
  float* out = (float*)d_out;

  float* wsf  = (float*)d_ws;
  float* w1v  = wsf;                 // 512 floats
  float* ctx  = wsf + Sn;            // 32*512 floats
  float* cvec = ctx + Bn * Sn;       // 32*512 floats

  const size_t need_split = (size_t)(Sn + 2 * Bn * Sn) * sizeof(float);

  k_w1v<<<dim3(16), dim3(256), 0, stream>>>(W1, V, w1v);
  k_attn_ctx<<<dim3(Bn), dim3(256), 0, stream>>>(x_static, w1v, ctx);

  if (ws_size >= need_split) {
    k_cvec<<<dim3(2, 2), dim3(128), 0, stream>>>(ctx, W3, b3, cvec);
    k_out_gemm_x<<<dim3((Bn * Tn) / 16, 2), dim3(128), 0, stream>>>(x, W3, cvec, out);
  } else {
    k_out_gemm_fused<<<dim3((Bn * Tn) / 16, 2), dim3(128), 0, stream>>>(x, ctx, W3, b3, out);
  }
}